// RNN_72902774882786
// MI455X (gfx1250) — compile-verified
//
#include <hip/hip_runtime.h>

// Leaky RNN scan for MI455X (gfx1250, wave32).
//   T=2048, B=64, N_IN=2, H=256
// Per step: s = 0.8*s + 0.2*(relu(s)@W_hh + x@W_ih + b + SIGMA*noise)
// Strategy:
//  - 4 workgroups, one per 16-row batch tile (rows are independent chains).
//  - 16 waves/WG, one 16-col tile of H each; W_hh B-fragments VGPR-resident.
//  - relu(state) double-buffered in LDS (2 x 16KB); one barrier per step.
//  - Exact fp32 matmul via V_WMMA_F32_16X16X4_F32, 4 interleaved accumulators.

typedef __attribute__((ext_vector_type(2))) float v2f;
typedef __attribute__((ext_vector_type(8))) float v8f;

#define T_STEPS 2048
#define B_SZ    64
#define H_SZ    256
#define SIGMA_F 0.15811388300841897f   // sqrt(2/0.2)*0.05
#define ONE_MINUS_ALPHA 0.8f
#define ALPHA_F 0.2f

__global__ __launch_bounds__(512, 1) void rnn_scan_kernel(
    const float* __restrict__ inputs,   // [T, B, 2]
    const float* __restrict__ h0,       // [B, H]
    const float* __restrict__ noise,    // [T, B, H]
    const float* __restrict__ wih,      // [2, H]
    const float* __restrict__ whh,      // [H, H]
    const float* __restrict__ bias,     // [1, H]
    float* __restrict__ out)            // [T+1, B, H]
{
    // relu(state) tile, double buffered: [buf][row 0..15][col 0..255]
    __shared__ float sbuf[2][16 * H_SZ];

    const int lane = threadIdx.x & 31;
    const int wave = threadIdx.x >> 5;     // 0..15 -> column tile
    const int lo   = lane & 15;
    const int hi   = lane >> 4;            // half-wave
    const int j    = wave * 16 + lo;       // owned column
    const int b0   = blockIdx.x * 16;      // batch-tile base row

    // Per-lane step-invariant scalars
    const float wih0 = wih[j];
    const float wih1 = wih[H_SZ + j];
    const float bj   = bias[j];

    // W_hh B-fragments, VGPR-resident (ISA B 4x16 layout:
    // VGPR0 = rows k, k+2 (per half-wave), VGPR1 = rows k+1, k+3, col = lo).
    v2f wB[64];
#pragma unroll
    for (int c = 0; c < 64; ++c) {
        const int k = 4 * c + 2 * hi;
        v2f w;
        w.x = whh[(size_t)k * H_SZ + j];
        w.y = whh[(size_t)(k + 1) * H_SZ + j];
        wB[c] = w;
    }

    // State registers follow the C/D 16x16 layout: reg r, lane -> row r+8*hi, col j.
    v8f s;
#pragma unroll
    for (int r = 0; r < 8; ++r) {
        const int m = r + 8 * hi;                 // local row 0..15
        const float v = h0[(size_t)(b0 + m) * H_SZ + j];
        s[r] = v;
        sbuf[0][m * H_SZ + j] = fmaxf(v, 0.0f);   // relu'd copy for the GEMM
        out[(size_t)(b0 + m) * H_SZ + j] = v;     // out[0] = initial_state
    }
    __syncthreads();

    // A-fragment base element offset in the LDS tile (ISA A 16x4 layout:
    // lanes 0-15 read S[lo][k..k+1], lanes 16-31 read S[lo][k+2..k+3]).
    const int aoff = lo * H_SZ + hi * 2;

#pragma unroll 1
    for (int t = 0; t < T_STEPS; ++t) {
        const float* rb = sbuf[t & 1];
        float*       wb = sbuf[(t & 1) ^ 1];

        // Per-step inputs: x (broadcast loads) and noise (coalesced).
        v2f   xv[8];
        float nz[8];
#pragma unroll
        for (int r = 0; r < 8; ++r) {
            const int m = r + 8 * hi;
            xv[r] = *(const v2f*)(inputs + ((size_t)t * B_SZ + b0 + m) * 2);
            nz[r] = noise[((size_t)t * B_SZ + b0 + m) * H_SZ + j];
        }
        // Hide HBM latency across the serial chain: prefetch next step's noise.
        if (t + 1 < T_STEPS) {
#pragma unroll
            for (int r = 0; r < 8; r += 2) {
                const int m = r + 8 * hi;
                __builtin_prefetch(
                    noise + ((size_t)(t + 1) * B_SZ + b0 + m) * H_SZ + j, 0, 0);
            }
        }

        // 16x256 * 256x16 fp32 GEMM: 64 chained WMMAs split over 4 accumulators.
        v8f acc[4] = {v8f{}, v8f{}, v8f{}, v8f{}};
#pragma unroll
        for (int c = 0; c < 64; ++c) {
            const v2f a = *(const v2f*)&rb[aoff + 4 * c];
            acc[c & 3] = __builtin_amdgcn_wmma_f32_16x16x4_f32(
                false, a, false, wB[c], (short)0, acc[c & 3], false, false);
        }
        const v8f g = (acc[0] + acc[1]) + (acc[2] + acc[3]);

        // Leaky update, publish relu(state) for next step, stream state out.
#pragma unroll
        for (int r = 0; r < 8; ++r) {
            const int m = r + 8 * hi;
            const float snew = g[r] + bj + xv[r].x * wih0 + xv[r].y * wih1
                             + SIGMA_F * nz[r];
            const float v = ONE_MINUS_ALPHA * s[r] + ALPHA_F * snew;
            s[r] = v;
            wb[m * H_SZ + j] = fmaxf(v, 0.0f);
            out[((size_t)(t + 1) * B_SZ + (b0 + m)) * H_SZ + j] = v;
        }
        __syncthreads();
    }
}

extern "C" void kernel_launch(void* const* d_in, const int* in_sizes, int n_in,
                              void* d_out, int out_size, void* d_ws, size_t ws_size,
                              hipStream_t stream) {
    (void)in_sizes; (void)n_in; (void)d_ws; (void)ws_size; (void)out_size;
    const float* inputs = (const float*)d_in[0];   // [T, B, 2]
    const float* h0     = (const float*)d_in[1];   // [B, H]
    const float* noise  = (const float*)d_in[2];   // [T, B, H]
    const float* wih    = (const float*)d_in[3];   // [2, H]
    const float* whh    = (const float*)d_in[4];   // [H, H]
    const float* bias   = (const float*)d_in[5];   // [1, H]
    float* out = (float*)d_out;                    // [T+1, B, H]

    rnn_scan_kernel<<<dim3(B_SZ / 16), dim3(512), 0, stream>>>(
        inputs, h0, noise, wih, whh, bias, out);
}